// DepthwiseSpatialConvROCm_25228637897546
// MI455X (gfx1250) — compile-verified
//
#include <hip/hip_runtime.h>
#include <math.h>

// Problem constants (from reference): B=32, I=16, C=128, T=2048, D=8, OUT_CH=128
#define NB   32
#define NI   16
#define NC   128
#define NT   2048
#define ND   8
#define NOUT 128   // NI*ND

typedef __attribute__((ext_vector_type(2))) float v2f;
typedef __attribute__((ext_vector_type(8))) float v8f;

// ---------------------------------------------------------------------------
// Kernel 1: max_norm renormalization of the (128 x 128) weight matrix.
// One block per output channel row; 128 threads reduce sum-of-squares in LDS.
// ---------------------------------------------------------------------------
__global__ __launch_bounds__(128)
void maxnorm_kernel(const float* __restrict__ w, float* __restrict__ wn) {
    __shared__ float red[128];
    const int row = blockIdx.x;      // 0..127 (out channel)
    const int tid = threadIdx.x;     // 0..127 (c index)
    const float v = w[row * NC + tid];
    red[tid] = v * v;
    __syncthreads();
    #pragma unroll
    for (int s = 64; s > 0; s >>= 1) {
        if (tid < s) red[tid] += red[tid + s];
        __syncthreads();
    }
    const float norm  = sqrtf(red[0]);
    const float scale = fminf(1.0f, 1.0f / fmaxf(norm, 1e-12f));
    wn[row * NC + tid] = v * scale;
}

// ---------------------------------------------------------------------------
// Kernel 2: out[b, i*8+d, t] = sum_c wn[i*8+d, c] * x[b,i,c,t]
// Block = one (b,i) and 256 t-values: 8 waves x (2 tiles of 16 t).
// Per wave: 64x V_WMMA_F32_16X16X4_F32 (two accumulator chains sharing A).
// A tile (16x4): rows 0..7 = weight, rows 8..15 = 0 (zero-padded in LDS).
// ---------------------------------------------------------------------------
__global__ __launch_bounds__(256)
void dwconv_wmma_kernel(const float* __restrict__ x,
                        const float* __restrict__ wn,
                        float* __restrict__ out) {
    // Padded stride 129 -> ds_load bank index (m*129 + c) % 64 = (m + c) % 64:
    // the 16 lanes of an A-fragment load hit 16 distinct banks (conflict-free).
    __shared__ float wA[16 * 129];

    const int tid = threadIdx.x;
    const int i   = blockIdx.y;      // in-channel group 0..15
    const int b   = blockIdx.z;      // batch 0..31

    // Stage A for this i: rows d=0..7 real, rows 8..15 zero.
    // k = r*256 + tid makes (d < 8) compile-time uniform per unrolled r:
    // r<=3 -> always a real row, r>=4 -> always the zero pad. Straight-line code.
    #pragma unroll
    for (int r = 0; r < 8; ++r) {
        const int k = r * 256 + tid;         // 0..2047
        const int d = k >> 7;                // 2r + (tid>>7)
        const int c = k & 127;
        float v = 0.0f;
        if (d < 8) v = wn[i * (8 * NC) + k]; // rows i*8 .. i*8+7, contiguous
        wA[d * 129 + c] = v;
    }
    __syncthreads();

    const int wave = tid >> 5;       // 0..7 (wave32)
    const int lane = tid & 31;
    const int m    = lane & 15;      // M row (A) / N col (B, t offset)
    const int hi   = lane >> 4;      // lane half selects K phase
    const int t0   = blockIdx.x * 256 + wave * 32;   // two 16-t tiles: t0, t0+16

    // x slice for (b,i): C x T row-major, element (c,t) at c*NT + t
    const float* __restrict__ xp =
        x + (size_t)(b * NI + i) * (size_t)(NC * NT);

    const int k_off0 = hi ? 2 : 0;   // B VGPR0 carries K=0 (lanes 0-15) / K=2 (16-31)
    const int k_off1 = hi ? 3 : 1;   // B VGPR1 carries K=1 / K=3

    v8f acc0 = {};
    v8f acc1 = {};
    #pragma unroll
    for (int g = 0; g < 32; ++g) {   // K = 128 in groups of 4
        const int c0 = g * 4;
        v2f a, b0f, b1f;
        // A fragment from LDS (16x4 f32 layout, ISA 7.12.2); consecutive
        // addresses -> single ds_load_b64 per group, shared by both tiles.
        a.x = wA[m * 129 + c0 + k_off0];
        a.y = wA[m * 129 + c0 + k_off1];
        // B fragments: each b32 load covers two contiguous 64B rows of x.
        const size_t row0 = (size_t)(c0 + k_off0) * NT;
        const size_t row1 = (size_t)(c0 + k_off1) * NT;
        b0f.x = xp[row0 + t0 + m];
        b0f.y = xp[row1 + t0 + m];
        b1f.x = xp[row0 + t0 + 16 + m];
        b1f.y = xp[row1 + t0 + 16 + m];
        // 8-arg form: (neg_a, A, neg_b, B, c_mod, C, reuse_a, reuse_b)
        acc0 = __builtin_amdgcn_wmma_f32_16x16x4_f32(
                   false, a, false, b0f, (short)0, acc0, false, false);
        acc1 = __builtin_amdgcn_wmma_f32_16x16x4_f32(
                   false, a, false, b1f, (short)0, acc1, false, false);
    }

    // D tiles: VGPR r = row d=r (lanes 0-15 valid), d=r+8 (lanes 16-31, discard)
    if (lane < 16) {
        float* __restrict__ op =
            out + ((size_t)b * NOUT + i * 8) * NT + t0 + lane;
        #pragma unroll
        for (int r = 0; r < 8; ++r) {
            op[(size_t)r * NT]      = acc0[r];
            op[(size_t)r * NT + 16] = acc1[r];
        }
    }
}

// ---------------------------------------------------------------------------
extern "C" void kernel_launch(void* const* d_in, const int* in_sizes, int n_in,
                              void* d_out, int out_size, void* d_ws, size_t ws_size,
                              hipStream_t stream) {
    const float* x = (const float*)d_in[0];   // (32,16,128,2048) f32
    const float* w = (const float*)d_in[1];   // (128,1,128,1)    f32
    float* out = (float*)d_out;               // (32,128,1,2048)  f32
    float* wn  = (float*)d_ws;                // 128*128 f32 = 64 KB scratch

    maxnorm_kernel<<<dim3(128), dim3(128), 0, stream>>>(w, wn);

    // grid: (T/256, I, B) = (8,16,32) blocks of 256 threads (8 waves)
    dwconv_wmma_kernel<<<dim3(NT / 256, NI, NB), dim3(256), 0, stream>>>(x, wn, out);
}